// GRUPolicyNetwork_46248207844034
// MI455X (gfx1250) — compile-verified
//
#include <hip/hip_runtime.h>

typedef float v2f __attribute__((ext_vector_type(2)));
typedef float v8f __attribute__((ext_vector_type(8)));

__device__ __forceinline__ float sigm_f(float x) { return 1.0f / (1.0f + __expf(-x)); }
__device__ __forceinline__ float tanh_f(float x) { return 2.0f / (1.0f + __expf(-2.0f * x)) - 1.0f; }

// ---- WMMA matvec, A chunks loaded from global (used once; critic path) ----
// y = W(ROWS x K, row-major ld=LDW) @ v via chained V_WMMA_F32_16X16X4_F32.
// B columns all hold v (broadcast), so D is y replicated across columns:
// lane (m = lane&15, kh = lane>>4) gets y[8*kh + j] in acc[j].
template <int ROWS, int LDW, int K>
__device__ __forceinline__ v8f wmma_mv16(const float* __restrict__ W,
                                         const float* __restrict__ vlds,
                                         int m, int kh) {
    v8f acc = {0.f, 0.f, 0.f, 0.f, 0.f, 0.f, 0.f, 0.f};
    const int   mi  = (ROWS >= 16) ? m : (m < ROWS ? m : 0);
    const float msk = (ROWS >= 16) ? 1.0f : (m < ROWS ? 1.0f : 0.0f);
#pragma unroll
    for (int c = 0; c < K; c += 4) {
        const int k0 = c + 2 * kh;
        v2f a, b;
        a.x = msk * W[mi * LDW + k0];
        a.y = msk * W[mi * LDW + k0 + 1];
        b.x = vlds[k0];
        b.y = vlds[k0 + 1];
        acc = __builtin_amdgcn_wmma_f32_16x16x4_f32(false, a, false, b,
                                                    (short)0, acc, false, false);
    }
    return acc;
}

// ---- WMMA matvec, A chunks preloaded in registers (actor recurrence path) ----
template <int NC>
__device__ __forceinline__ v8f wmma_mv_reg(const v2f* A, const float* __restrict__ vlds,
                                           int kh) {
    v8f acc = {0.f, 0.f, 0.f, 0.f, 0.f, 0.f, 0.f, 0.f};
#pragma unroll
    for (int c = 0; c < NC; ++c) {
        const int k0 = 4 * c + 2 * kh;
        v2f b;
        b.x = vlds[k0];
        b.y = vlds[k0 + 1];
        acc = __builtin_amdgcn_wmma_f32_16x16x4_f32(false, A[c], false, b,
                                                    (short)0, acc, false, false);
    }
    return acc;
}

// Result is replicated across the 16 lanes of each half: lanes 0 and 16 publish it.
__device__ __forceinline__ void store16(float* dst, v8f y, int lane) {
    if ((lane & 15) == 0) {
        const int base = (lane >> 4) * 8;
#pragma unroll
        for (int j = 0; j < 8; ++j) dst[base + j] = y[j];
    }
}

#define NSTEP 16

__global__ __launch_bounds__(32) void gru_policy_kernel(
    const float* __restrict__ features,
    const float* __restrict__ W_emb, const float* __restrict__ b_emb,
    const float* __restrict__ W_ih,  const float* __restrict__ b_ih,
    const float* __restrict__ W_hh,  const float* __restrict__ b_hh,
    const float* __restrict__ W_out, const float* __restrict__ b_out,
    const float* __restrict__ W_v1,  const float* __restrict__ b_v1,
    const float* __restrict__ W_v2,  const float* __restrict__ b_v2,
    const float* __restrict__ W_v3,  const float* __restrict__ b_v3,
    const float* __restrict__ W_v4,  const float* __restrict__ b_v4,
    float* __restrict__ out) {
    __shared__ float s_x[4];     // decoder input (previous logits)
    __shared__ float s_e[16];    // embedded input
    __shared__ float s_h[16];    // hidden state (scalar view for B construction)
    __shared__ float s_va[128];  // critic ping
    __shared__ float s_vb[128];  // critic pong

    const int lane = threadIdx.x;
    const int m = lane & 15;
    const int kh = lane >> 4;

    if (blockIdx.x == 0) {
        // ---------------- actor: 16 sequential GRU decoder steps ----------------
        if (lane < 4) s_x[lane] = 0.0f;
        if (lane < 16) s_h[lane] = features[lane];
        __syncthreads();

        v8f h;
#pragma unroll
        for (int j = 0; j < 8; ++j) h[j] = features[8 * kh + j];

        // ---- preload ALL loop-invariant weights into registers (A-layout chunks) ----
        // A chunk layout: lane holds W[m][4c + 2*kh + {0,1}]
        v2f aemb;                       // W_emb: 16x4, one chunk
        aemb.x = W_emb[m * 4 + 2 * kh];
        aemb.y = W_emb[m * 4 + 2 * kh + 1];

        v2f aih[3][4], ahh[3][4];       // gate blocks r,z,n: 16x16 each, 4 chunks each
#pragma unroll
        for (int g = 0; g < 3; ++g)
#pragma unroll
            for (int c = 0; c < 4; ++c) {
                const int row = (g * 16 + m) * 16;
                const int k0 = 4 * c + 2 * kh;
                aih[g][c].x = W_ih[row + k0];
                aih[g][c].y = W_ih[row + k0 + 1];
                ahh[g][c].x = W_hh[row + k0];
                ahh[g][c].y = W_hh[row + k0 + 1];
            }

        v2f aout[4];                    // W_out: 4x16, zero-padded to 16 rows
        {
            const float msk = (m < 4) ? 1.0f : 0.0f;
            const int mi = (m < 4) ? m : 0;
#pragma unroll
            for (int c = 0; c < 4; ++c) {
                const int k0 = 4 * c + 2 * kh;
                aout[c].x = msk * W_out[mi * 16 + k0];
                aout[c].y = msk * W_out[mi * 16 + k0 + 1];
            }
        }

        // biases in replicated-row layout (index 8*kh + j)
        float bemb[8], bir[8], biz[8], bin_[8], bhr[8], bhz[8], bhn[8];
#pragma unroll
        for (int j = 0; j < 8; ++j) {
            const int i = 8 * kh + j;
            bemb[j] = b_emb[i];
            bir[j] = b_ih[i];       bhr[j] = b_hh[i];
            biz[j] = b_ih[16 + i];  bhz[j] = b_hh[16 + i];
            bin_[j] = b_ih[32 + i]; bhn[j] = b_hh[32 + i];
        }
        const float bo0 = b_out[0], bo1 = b_out[1], bo2 = b_out[2], bo3 = b_out[3];

        int target_len = 0;
        for (int step = 0; step < NSTEP; ++step) {
            // e = relu(W_emb @ x + b_emb)          (one 16x16x4 WMMA)
            v8f e = wmma_mv_reg<1>(&aemb, s_x, kh);
#pragma unroll
            for (int j = 0; j < 8; ++j) {
                const float t = e[j] + bemb[j];
                e[j] = t > 0.0f ? t : 0.0f;
            }
            store16(s_e, e, lane);
            __syncthreads();

            // six 16x16 matvecs, each a chain of 4 register-A WMMAs
            v8f gxr = wmma_mv_reg<4>(aih[0], s_e, kh);
            v8f gxz = wmma_mv_reg<4>(aih[1], s_e, kh);
            v8f gxn = wmma_mv_reg<4>(aih[2], s_e, kh);
            v8f ghr = wmma_mv_reg<4>(ahh[0], s_h, kh);
            v8f ghz = wmma_mv_reg<4>(ahh[1], s_h, kh);
            v8f ghn = wmma_mv_reg<4>(ahh[2], s_h, kh);

            v8f hn;
#pragma unroll
            for (int j = 0; j < 8; ++j) {
                const float r = sigm_f((gxr[j] + bir[j]) + (ghr[j] + bhr[j]));
                const float z = sigm_f((gxz[j] + biz[j]) + (ghz[j] + bhz[j]));
                const float n = tanh_f((gxn[j] + bin_[j]) + r * (ghn[j] + bhn[j]));
                hn[j] = (1.0f - z) * n + z * h[j];
            }
            h = hn;
            __syncthreads();              // old s_h readers done before overwrite
            store16(s_h, hn, lane);
            __syncthreads();

            // logits = W_out(padded) @ hn + b_out  (chain of 4 register-A WMMAs)
            v8f lp = wmma_mv_reg<4>(aout, s_h, kh);
            const float l0 = lp[0] + bo0, l1 = lp[1] + bo1;
            const float l2 = lp[2] + bo2, l3 = lp[3] + bo3;   // valid on kh==0 lanes

            if (lane == 0) {
                s_x[0] = l0; s_x[1] = l1; s_x[2] = l2; s_x[3] = l3;
                float mx = fmaxf(fmaxf(l0, l1), fmaxf(l2, l3));
                float e0 = __expf(l0 - mx), e1 = __expf(l1 - mx);
                float e2 = __expf(l2 - mx), e3 = __expf(l3 - mx);
                const float inv = 1.0f / (e0 + e1 + e2 + e3);
                e0 *= inv; e1 *= inv; e2 *= inv; e3 *= inv;
                if (step == 1)
                    target_len = (int)(rintf(e0 * 2.0f) + rintf(e1 * 5.0f) + rintf(e2 * 7.0f));
                const float keep = (step < 2) ? 1.0f
                                  : (((step - 2) < target_len) ? 1.0f : 0.0f);
                out[step * 4 + 0] = e0 * keep;
                out[step * 4 + 1] = e1 * keep;
                out[step * 4 + 2] = e2 * keep;
                out[step * 4 + 3] = e3 * keep;
            }
            __syncthreads();
        }
    } else {
        // ---------------- critic: 16 -> 128 -> 64 -> 32 -> 32 MLP ----------------
        if (lane < 16) s_va[lane] = features[lane];
        __syncthreads();

        // layer 1: 128 outputs = 8 row-blocks, K=16
        for (int bi = 0; bi < 8; ++bi) {
            v8f acc = wmma_mv16<16, 16, 16>(W_v1 + bi * 256, s_va, m, kh);
#pragma unroll
            for (int j = 0; j < 8; ++j) {
                const float t = acc[j] + b_v1[bi * 16 + 8 * kh + j];
                acc[j] = t > 0.0f ? t : 0.0f;
            }
            store16(s_vb + bi * 16, acc, lane);
        }
        __syncthreads();

        // layer 2: 64 outputs, K=128
        for (int bi = 0; bi < 4; ++bi) {
            v8f acc = wmma_mv16<16, 128, 128>(W_v2 + bi * 2048, s_vb, m, kh);
#pragma unroll
            for (int j = 0; j < 8; ++j) {
                const float t = acc[j] + b_v2[bi * 16 + 8 * kh + j];
                acc[j] = t > 0.0f ? t : 0.0f;
            }
            store16(s_va + bi * 16, acc, lane);
        }
        __syncthreads();

        // layer 3: 32 outputs, K=64
        for (int bi = 0; bi < 2; ++bi) {
            v8f acc = wmma_mv16<16, 64, 64>(W_v3 + bi * 1024, s_va, m, kh);
#pragma unroll
            for (int j = 0; j < 8; ++j) {
                const float t = acc[j] + b_v3[bi * 16 + 8 * kh + j];
                acc[j] = t > 0.0f ? t : 0.0f;
            }
            store16(s_vb + bi * 16, acc, lane);
        }
        __syncthreads();

        // layer 4: 32 outputs, K=32
        for (int bi = 0; bi < 2; ++bi) {
            v8f acc = wmma_mv16<16, 32, 32>(W_v4 + bi * 512, s_vb, m, kh);
#pragma unroll
            for (int j = 0; j < 8; ++j) {
                const float t = acc[j] + b_v4[bi * 16 + 8 * kh + j];
                acc[j] = t > 0.0f ? t : 0.0f;
            }
            store16(s_va + bi * 16, acc, lane);
        }
        __syncthreads();

        out[64 + lane] = s_va[lane];
    }
}

extern "C" void kernel_launch(void* const* d_in, const int* in_sizes, int n_in,
                              void* d_out, int out_size, void* d_ws, size_t ws_size,
                              hipStream_t stream) {
    (void)in_sizes; (void)n_in; (void)out_size; (void)d_ws; (void)ws_size;
    const float* features = (const float*)d_in[0];
    const float* W_emb = (const float*)d_in[1];
    const float* b_emb = (const float*)d_in[2];
    const float* W_ih  = (const float*)d_in[3];
    const float* b_ih  = (const float*)d_in[4];
    const float* W_hh  = (const float*)d_in[5];
    const float* b_hh  = (const float*)d_in[6];
    const float* W_out = (const float*)d_in[7];
    const float* b_out = (const float*)d_in[8];
    const float* W_v1  = (const float*)d_in[9];
    const float* b_v1  = (const float*)d_in[10];
    const float* W_v2  = (const float*)d_in[11];
    const float* b_v2  = (const float*)d_in[12];
    const float* W_v3  = (const float*)d_in[13];
    const float* b_v3  = (const float*)d_in[14];
    const float* W_v4  = (const float*)d_in[15];
    const float* b_v4  = (const float*)d_in[16];

    // block 0 = actor wave (serial GRU), block 1 = critic wave (MLP), run concurrently
    gru_policy_kernel<<<dim3(2), dim3(32), 0, stream>>>(
        features, W_emb, b_emb, W_ih, b_ih, W_hh, b_hh, W_out, b_out,
        W_v1, b_v1, W_v2, b_v2, W_v3, b_v3, W_v4, b_v4, (float*)d_out);
}